// MLAHybridLoop12MonarchAttnLoRAFfnBinaryDPTransformer_51041391346394
// MI455X (gfx1250) — compile-verified
//
#include <hip/hip_runtime.h>
#include <hip/hip_bf16.h>

// ---------------------------------------------------------------------------
// PKM layer on MI455X (gfx1250, wave32):
//   K1: q_down = x @ qd_w.T + qd_b           (WMMA f32 16x16x4, 64x32/wave)
//   K2: q      = q_down @ qu_w.T             (WMMA f32 16x16x4, 64x32/wave)
//   K3: per-(token,head) bucket scores + sorted-merge trellis top-32 + softmax
//   K4: weighted gather over values table    (memory-roofline stage, ~537 MB)
//   K5: y      = out_v @ vp_w.T              (WMMA f32 16x16x4, 64x32/wave)
// ---------------------------------------------------------------------------

#define N_TOK   2048        // BATCH * SEQ
#define D_MODEL 1024
#define Q_RANK  512
#define H       4
#define KEY_DIM 288
#define HKD     (H * KEY_DIM)   // 1152
#define VDIM    512
#define NBUCK   18
#define BDIM    16
#define KNN     32
#define KPT     (H * KNN)       // 128 picks per token

typedef __attribute__((ext_vector_type(2))) float v2f;
typedef __attribute__((ext_vector_type(8))) float v8f;

// ---------------------------------------------------------------------------
// WMMA GEMM: C[M,N] = A[M,K] @ W[N,K]^T (+ bias[N] if non-null)
// One wave -> 64x32 output tile = 4x2 grid of 16x16 WMMA tiles:
//   8 independent accumulation chains (hides f32-XDL latency at ~1 wave/SIMD)
//   fed by 6 b64 loads per K-step (4 A-frags shared across 2 N-tiles,
//   2 B-frags shared across 4 M-tiles).
// f32 A fragment (16x4): lane = (k>=2)*16 + m, vgpr = k%2
//   -> per lane: float2 at A[(m0+lane%16)*K + k + (lane/16)*2]
// f32 B fragment (4x16) mirrors A with N in place of M.
// C/D: vgpr v, lanes 0-15 -> M=v, N=lane; lanes 16-31 -> M=8+v, N=lane-16.
// ---------------------------------------------------------------------------
__global__ void __launch_bounds__(128)
wmma_gemm_xwt_kernel(const float* __restrict__ A,
                     const float* __restrict__ W,
                     const float* __restrict__ bias,
                     float* __restrict__ C,
                     int M, int N, int K) {
  const int wave = (blockIdx.x * blockDim.x + threadIdx.x) >> 5;
  const int lane = threadIdx.x & 31;
  const int Nt = N >> 5;                       // 32-wide tiles in N
  const int Mt = M >> 6;                       // 64-tall tiles in M
  if (wave >= Mt * Nt) return;
  const int m0 = (wave / Nt) << 6;
  const int n0 = (wave % Nt) << 5;

  const int mr = lane & 15;                    // row within 16-tile (M or N)
  const int kh = (lane >> 4) << 1;             // K sub-offset: 0 or 2

  const float* aR0 = A + (size_t)(m0 + mr) * K + kh;
  const float* aR1 = aR0 + (size_t)16 * K;
  const float* aR2 = aR0 + (size_t)32 * K;
  const float* aR3 = aR0 + (size_t)48 * K;
  const float* bR0 = W + (size_t)(n0 + mr) * K + kh;
  const float* bR1 = bR0 + (size_t)16 * K;

  v8f acc[4][2] = {{{}, {}}, {{}, {}}, {{}, {}}, {{}, {}}};

  for (int k = 0; k < K; k += 4) {
    v2f a0 = *(const v2f*)(aR0 + k);
    v2f a1 = *(const v2f*)(aR1 + k);
    v2f a2 = *(const v2f*)(aR2 + k);
    v2f a3 = *(const v2f*)(aR3 + k);
    v2f b0 = *(const v2f*)(bR0 + k);
    v2f b1 = *(const v2f*)(bR1 + k);
    acc[0][0] = __builtin_amdgcn_wmma_f32_16x16x4_f32(false, a0, false, b0,
                                                      (short)0, acc[0][0], false, false);
    acc[1][0] = __builtin_amdgcn_wmma_f32_16x16x4_f32(false, a1, false, b0,
                                                      (short)0, acc[1][0], false, false);
    acc[2][0] = __builtin_amdgcn_wmma_f32_16x16x4_f32(false, a2, false, b0,
                                                      (short)0, acc[2][0], false, false);
    acc[3][0] = __builtin_amdgcn_wmma_f32_16x16x4_f32(false, a3, false, b0,
                                                      (short)0, acc[3][0], false, false);
    acc[0][1] = __builtin_amdgcn_wmma_f32_16x16x4_f32(false, a0, false, b1,
                                                      (short)0, acc[0][1], false, false);
    acc[1][1] = __builtin_amdgcn_wmma_f32_16x16x4_f32(false, a1, false, b1,
                                                      (short)0, acc[1][1], false, false);
    acc[2][1] = __builtin_amdgcn_wmma_f32_16x16x4_f32(false, a2, false, b1,
                                                      (short)0, acc[2][1], false, false);
    acc[3][1] = __builtin_amdgcn_wmma_f32_16x16x4_f32(false, a3, false, b1,
                                                      (short)0, acc[3][1], false, false);
  }

  const int col = mr;
  const int rowBase = (lane >> 4) << 3;        // 0 or 8
  const float bv0 = bias ? bias[n0 + col] : 0.f;
  const float bv1 = bias ? bias[n0 + 16 + col] : 0.f;

#pragma unroll
  for (int mi = 0; mi < 4; ++mi) {
#pragma unroll
    for (int v = 0; v < 8; ++v) {
      const int row = m0 + mi * 16 + rowBase + v;
      C[(size_t)row * N + n0 + col]      = acc[mi][0][v] + bv0;
      C[(size_t)row * N + n0 + 16 + col] = acc[mi][1][v] + bv1;
    }
  }
}

// ---------------------------------------------------------------------------
// Trellis search + softmax. One thread per (token, head) row (8192 rows).
// Since penalties stay sorted ascending and delta >= 0, each top_k step of
// concat(P, P+delta) is a stable 2-way merge of sorted lists, take first k.
// ---------------------------------------------------------------------------
__global__ void pkm_trellis_kernel(const float* __restrict__ q,
                                   const float* __restrict__ keys,
                                   int* __restrict__ idx_out,
                                   float* __restrict__ w_out) {
  const int r = blockIdx.x * blockDim.x + threadIdx.x;   // r = n*H + h
  if (r >= N_TOK * H) return;
  const int n = r >> 2;
  const int h = r & 3;

  const float* qrow = q + (size_t)n * HKD + h * KEY_DIM;

  float deltas[NBUCK];
  int best_code = 0;
#pragma unroll
  for (int m = 0; m < NBUCK; ++m) {
    const float* kq = qrow + m * BDIM;
    const float* k0 = keys + (size_t)((h * NBUCK + m) * 2) * BDIM;
    const float* k1 = k0 + BDIM;
    float s0 = 0.f, s1 = 0.f;
#pragma unroll
    for (int d = 0; d < BDIM; ++d) {
      s0 = fmaf(kq[d], k0[d], s0);
      s1 = fmaf(kq[d], k1[d], s1);
    }
    if (s1 > s0) best_code |= (1 << m);
    deltas[m] = fabsf(s0 - s1);
  }

  float pen[KNN];
  int msk[KNN];
  pen[0] = 0.f;
  msk[0] = 0;
  int cur = 1;
  for (int t = 0; t < NBUCK; ++t) {
    const float dl = deltas[t];
    const int bw = 1 << t;
    int k = cur * 2;
    if (k > KNN) k = KNN;
    float np[KNN];
    int nm[KNN];
    int i = 0, j = 0;
    for (int o = 0; o < k; ++o) {
      // stable: prefer list A (original penalties, lower concat index) on ties
      const bool takeA = (j >= cur) || ((i < cur) && (pen[i] <= pen[j] + dl));
      if (takeA) { np[o] = pen[i]; nm[o] = msk[i]; ++i; }
      else       { np[o] = pen[j] + dl; nm[o] = msk[j] ^ bw; ++j; }
    }
    for (int o = 0; o < k; ++o) { pen[o] = np[o]; msk[o] = nm[o]; }
    cur = k;
  }

  // softmax over (best_score - pen[o]); best_score cancels, pen[0] is max score
  float e[KNN];
  float sum = 0.f;
#pragma unroll
  for (int o = 0; o < KNN; ++o) {
    e[o] = __expf(pen[0] - pen[o]);
    sum += e[o];
  }
  const float inv = 1.f / sum;

  int* io = idx_out + (size_t)n * KPT + h * KNN;
  float* wo = w_out + (size_t)n * KPT + h * KNN;
#pragma unroll
  for (int o = 0; o < KNN; ++o) {
    io[o] = best_code ^ msk[o];
    wo[o] = e[o] * inv;
  }
}

// ---------------------------------------------------------------------------
// Weighted gather (embedding_bag sum) -- the HBM-roofline stage (~537 MB).
// One block per token; (idx, w) staged in LDS; each of 256 threads owns 2 of
// the 512 value dims -> coalesced float2 loads of the random value rows.
// 4-deep software pipeline: 4 independent row loads in flight per group, two
// accumulator pairs for FMA ILP, one prefetch per group at distance 8.
// ---------------------------------------------------------------------------
__global__ void __launch_bounds__(256)
pkm_gather_kernel(const int* __restrict__ idx,
                  const float* __restrict__ w,
                  const float* __restrict__ values,
                  float* __restrict__ out_v) {
  __shared__ int s_idx[KPT];
  __shared__ float s_w[KPT];
  const int n = blockIdx.x;
  const int t = threadIdx.x;
  if (t < KPT) {
    s_idx[t] = idx[(size_t)n * KPT + t];
    s_w[t] = w[(size_t)n * KPT + t];
  }
  __syncthreads();

  const int d = t * 2;
  float2 accA = make_float2(0.f, 0.f);
  float2 accB = make_float2(0.f, 0.f);
  for (int k = 0; k < KPT; k += 4) {
    if (k + 8 < KPT)
      __builtin_prefetch(values + (size_t)s_idx[k + 8] * VDIM + d, 0, 0);
    const float2 v0 = *(const float2*)(values + (size_t)s_idx[k + 0] * VDIM + d);
    const float2 v1 = *(const float2*)(values + (size_t)s_idx[k + 1] * VDIM + d);
    const float2 v2 = *(const float2*)(values + (size_t)s_idx[k + 2] * VDIM + d);
    const float2 v3 = *(const float2*)(values + (size_t)s_idx[k + 3] * VDIM + d);
    const float w0 = s_w[k + 0];
    const float w1 = s_w[k + 1];
    const float w2 = s_w[k + 2];
    const float w3 = s_w[k + 3];
    accA.x = fmaf(w0, v0.x, accA.x);
    accA.y = fmaf(w0, v0.y, accA.y);
    accB.x = fmaf(w1, v1.x, accB.x);
    accB.y = fmaf(w1, v1.y, accB.y);
    accA.x = fmaf(w2, v2.x, accA.x);
    accA.y = fmaf(w2, v2.y, accA.y);
    accB.x = fmaf(w3, v3.x, accB.x);
    accB.y = fmaf(w3, v3.y, accB.y);
  }
  float2 acc = make_float2(accA.x + accB.x, accA.y + accB.y);
  *(float2*)(out_v + (size_t)n * VDIM + d) = acc;
}

// ---------------------------------------------------------------------------
extern "C" void kernel_launch(void* const* d_in, const int* in_sizes, int n_in,
                              void* d_out, int out_size, void* d_ws, size_t ws_size,
                              hipStream_t stream) {
  const float* x      = (const float*)d_in[0];  // [2048, 1024]
  const float* keys   = (const float*)d_in[1];  // [4, 18, 2, 16]
  const float* qd_w   = (const float*)d_in[2];  // [512, 1024]
  const float* qd_b   = (const float*)d_in[3];  // [512]
  const float* qu_w   = (const float*)d_in[4];  // [1152, 512]
  const float* values = (const float*)d_in[5];  // [262144, 512]
  const float* vp_w   = (const float*)d_in[6];  // [1024, 512]
  float* out = (float*)d_out;                   // [2048, 1024]

  // workspace layout
  float* q_down = (float*)d_ws;                         // 2048*512
  float* q      = q_down + (size_t)N_TOK * Q_RANK;      // 2048*1152
  int*   w_idx  = (int*)(q + (size_t)N_TOK * HKD);      // 2048*128
  float* w_wt   = (float*)(w_idx + (size_t)N_TOK * KPT);// 2048*128
  float* out_v  = w_wt + (size_t)N_TOK * KPT;           // 2048*512

  // K1: q_down = x @ qd_w.T + qd_b   (tiles = 32*16 = 512 -> 128 blocks)
  {
    int tiles = (N_TOK >> 6) * (Q_RANK >> 5);
    wmma_gemm_xwt_kernel<<<tiles / 4, 128, 0, stream>>>(
        x, qd_w, qd_b, q_down, N_TOK, Q_RANK, D_MODEL);
  }
  // K2: q = q_down @ qu_w.T          (tiles = 32*36 = 1152 -> 288 blocks)
  {
    int tiles = (N_TOK >> 6) * (HKD >> 5);
    wmma_gemm_xwt_kernel<<<tiles / 4, 128, 0, stream>>>(
        q_down, qu_w, nullptr, q, N_TOK, HKD, Q_RANK);
  }
  // K3: trellis + softmax            (8192 rows)
  pkm_trellis_kernel<<<(N_TOK * H) / 128, 128, 0, stream>>>(
      q, keys, w_idx, w_wt);
  // K4: weighted gather              (one block per token)
  pkm_gather_kernel<<<N_TOK, 256, 0, stream>>>(w_idx, w_wt, values, out_v);
  // K5: out = out_v @ vp_w.T         (tiles = 32*32 = 1024 -> 256 blocks)
  {
    int tiles = (N_TOK >> 6) * (D_MODEL >> 5);
    wmma_gemm_xwt_kernel<<<tiles / 4, 128, 0, stream>>>(
        out_v, vp_w, nullptr, out, N_TOK, D_MODEL, Q_RANK);
  }
}